// LeafFeatureExtractor_60876866453855
// MI455X (gfx1250) — compile-verified
//
#include <hip/hip_runtime.h>
#include <math.h>

// ---------------- problem constants (match reference) ----------------
#define BATCH     2
#define NPTS      8192
#define FEAT_DIM  64
#define DIN       70          // 64 features + 6 geometry
#define KPAD      72          // DIN rounded up to multiple of 4 for WMMA K-loop
#define HID       128
#define KNN       10
#define R2        (0.02f * 0.02f)

#define ASTRIDE   73          // odd stride -> conflict-free LDS column reads
#define HSTRIDE   129

// geometry kernel decomposition: 64 points x 4 j-segments per 256-thread block
#define GPTS      64
#define GSEG      4

typedef __attribute__((ext_vector_type(2))) float v2f;
typedef __attribute__((ext_vector_type(8))) float v8f;

// =====================================================================
// Kernel 1: per-batch masked centroid (sum.xyz + count) -> ws[b*4 + 0..3]
// =====================================================================
__global__ __launch_bounds__(256)
void leaf_center_kernel(const float* __restrict__ points,
                        const unsigned char* __restrict__ mask,
                        float* __restrict__ center_ws)
{
    const int b   = blockIdx.x;
    const int tid = threadIdx.x;
    const float* p = points + (size_t)b * NPTS * 3;
    const unsigned char* m = mask + (size_t)b * NPTS;

    float sx = 0.f, sy = 0.f, sz = 0.f, sc = 0.f;
    for (int i = tid; i < NPTS; i += 256) {
        float mv = m[i] ? 1.0f : 0.0f;
        sx += p[i * 3 + 0] * mv;
        sy += p[i * 3 + 1] * mv;
        sz += p[i * 3 + 2] * mv;
        sc += mv;
    }

    __shared__ float red[4][256];
    red[0][tid] = sx; red[1][tid] = sy; red[2][tid] = sz; red[3][tid] = sc;
    __syncthreads();
    for (int s = 128; s > 0; s >>= 1) {
        if (tid < s) {
            red[0][tid] += red[0][tid + s];
            red[1][tid] += red[1][tid + s];
            red[2][tid] += red[2][tid + s];
            red[3][tid] += red[3][tid + s];
        }
        __syncthreads();
    }
    if (tid == 0) {
        center_ws[b * 4 + 0] = red[0][0];
        center_ws[b * 4 + 1] = red[1][0];
        center_ws[b * 4 + 2] = red[2][0];
        center_ws[b * 4 + 3] = red[3][0];
    }
}

// =====================================================================
// Kernel 2: O(N^2) geometry, segment-parallel.
// Block = 64 points x 4 j-segments (256 threads, 8 waves). All threads
// cooperatively stage 256-point j-tiles in LDS; thread (il,seg) scans its
// 64-entry quarter of every tile, keeping a private partial top-10 (with
// neighbor coords in registers) + partial density. Partials merge via LDS;
// seg==0 threads finish covariance eigensolve + feature write.
// =====================================================================
__global__ __launch_bounds__(256)
void leaf_geom_kernel(const float* __restrict__ points,
                      const unsigned char* __restrict__ mask,
                      const float* __restrict__ center_ws,
                      float* __restrict__ geom_ws)
{
    const int b   = blockIdx.y;
    const int tid = threadIdx.x;
    const int il  = tid & (GPTS - 1);   // local point index   0..63
    const int seg = tid >> 6;           // j-segment           0..3
    const int i   = blockIdx.x * GPTS + il;

    const float* p = points + (size_t)b * NPTS * 3;
    const unsigned char* msk = mask + (size_t)b * NPTS;

    const float pix = p[i * 3 + 0];
    const float piy = p[i * 3 + 1];
    const float piz = p[i * 3 + 2];

    // partial top-KNN: squared distances (monotone in dist) + neighbor coords
    float knd[KNN];
    float knx[KNN], kny[KNN], knz[KNN];
#pragma unroll
    for (int t = 0; t < KNN; ++t) { knd[t] = 3.4e38f; knx[t] = kny[t] = knz[t] = 0.f; }

    float density = 0.0f;

    __shared__ float4 spts[256];             // x, y, z, mask
    __shared__ float4 mergebuf[256][KNN];    // per-thread partial lists (40KB)
    __shared__ float  densbuf[256];

    const int tlo = seg * 64, thi = tlo + 64;
    for (int base = 0; base < NPTS; base += 256) {
        {
            int j = base + tid;
            float4 v;
            v.x = p[j * 3 + 0]; v.y = p[j * 3 + 1]; v.z = p[j * 3 + 2];
            v.w = msk[j] ? 1.0f : 0.0f;
            spts[tid] = v;
        }
        __syncthreads();
#pragma unroll 4
        for (int t = tlo; t < thi; ++t) {
            float4 q = spts[t];
            float dx = q.x - pix, dy = q.y - piy, dz = q.z - piz;
            float d2 = fmaf(dx, dx, fmaf(dy, dy, dz * dz));
            density += ((d2 < R2) && (q.w != 0.0f)) ? 1.0f : 0.0f;
            if (d2 < knd[KNN - 1]) {            // candidate for partial top-10
                float dd = d2, px_ = q.x, py_ = q.y, pz_ = q.z;
#pragma unroll
                for (int s = 0; s < KNN; ++s) {  // sorted-insert swap chain
                    if (dd < knd[s]) {
                        float td = knd[s]; knd[s] = dd; dd = td;
                        float tx = knx[s]; knx[s] = px_; px_ = tx;
                        float ty = kny[s]; kny[s] = py_; py_ = ty;
                        float tz = knz[s]; knz[s] = pz_; pz_ = tz;
                    }
                }
            }
        }
        __syncthreads();
    }

    // publish partial lists + densities
#pragma unroll
    for (int t = 0; t < KNN; ++t)
        mergebuf[tid][t] = make_float4(knd[t], knx[t], kny[t], knz[t]);
    densbuf[tid] = density;
    __syncthreads();

    if (seg != 0) return;

    // merge 3 other segments' partials into this thread's list
    density += densbuf[il + 64] + densbuf[il + 128] + densbuf[il + 192];
#pragma unroll
    for (int s2 = 1; s2 < GSEG; ++s2) {
#pragma unroll
        for (int t = 0; t < KNN; ++t) {
            float4 c = mergebuf[s2 * 64 + il][t];
            if (c.x < knd[KNN - 1]) {
                float dd = c.x, px_ = c.y, py_ = c.z, pz_ = c.w;
#pragma unroll
                for (int s = 0; s < KNN; ++s) {
                    if (dd < knd[s]) {
                        float td = knd[s]; knd[s] = dd; dd = td;
                        float tx = knx[s]; knx[s] = px_; px_ = tx;
                        float ty = kny[s]; kny[s] = py_; py_ = ty;
                        float tz = knz[s]; knz[s] = pz_; pz_ = tz;
                    }
                }
            }
        }
    }

    // covariance of (neighbor - p_i), divided by KNN
    float cxx = 0.f, cyy = 0.f, czz = 0.f, cxy = 0.f, cxz = 0.f, cyz = 0.f;
#pragma unroll
    for (int t = 0; t < KNN; ++t) {
        float dx = knx[t] - pix, dy = kny[t] - piy, dz = knz[t] - piz;
        cxx += dx * dx; cyy += dy * dy; czz += dz * dz;
        cxy += dx * dy; cxz += dx * dz; cyz += dy * dz;
    }
    const float inv_k = 1.0f / (float)KNN;
    cxx *= inv_k; cyy *= inv_k; czz *= inv_k;
    cxy *= inv_k; cxz *= inv_k; cyz *= inv_k;

    // closed-form eigenvalues of symmetric 3x3 (trigonometric method)
    float qv = (cxx + cyy + czz) * (1.0f / 3.0f);
    float aq = cxx - qv, bq = cyy - qv, cq = czz - qv;
    float p1 = cxy * cxy + cxz * cxz + cyz * cyz;
    float p2 = aq * aq + bq * bq + cq * cq + 2.0f * p1;
    float pp = sqrtf(fmaxf(p2, 0.0f) * (1.0f / 6.0f));
    float ev0, ev2;
    if (pp > 1e-12f) {
        float det = aq * (bq * cq - cyz * cyz)
                  - cxy * (cxy * cq - cyz * cxz)
                  + cxz * (cxy * cyz - bq * cxz);
        float r = det / (2.0f * pp * pp * pp);
        r = fminf(1.0f, fmaxf(-1.0f, r));
        float phi = acosf(r) * (1.0f / 3.0f);
        ev2 = qv + 2.0f * pp * cosf(phi);                     // largest
        ev0 = qv + 2.0f * pp * cosf(phi + 2.09439510239f);    // smallest
    } else {
        ev0 = ev2 = qv;
    }
    float curv = ev0 / (ev2 + 1e-8f);

    // global-frame features relative to masked centroid
    float cnt   = center_ws[b * 4 + 3];
    float inv_c = 1.0f / fmaxf(cnt, 1.0f);
    float cx = center_ws[b * 4 + 0] * inv_c;
    float cy = center_ws[b * 4 + 1] * inv_c;
    float cz = center_ws[b * 4 + 2] * inv_c;
    float rx = pix - cx, ry = piy - cy, rz = piz - cz;

    float g0 = sqrtf(rx * rx + ry * ry + rz * rz);  // dist to center
    float g1 = rz;                                  // height
    float g2 = sqrtf(rx * rx + ry * ry);            // horizontal dist
    float g5 = atan2f(ry, rx);                      // azimuth

    float any = (cnt > 0.0f) ? 1.0f : 0.0f;
    float* g = geom_ws + ((size_t)b * NPTS + i) * 6;
    g[0] = g0 * any;
    g[1] = g1 * any;
    g[2] = g2 * any;
    g[3] = density * any;
    g[4] = curv * any;
    g[5] = g5 * any;
}

// =====================================================================
// Kernel 3: fused 2-layer MLP via V_WMMA_F32_16X16X4_F32 (wave32).
// One workgroup (8 waves) = one 16-row tile; wave w owns output cols
// [16w, 16w+16). A/H activations staged in LDS with odd strides.
// =====================================================================
__global__ __launch_bounds__(256)
void leaf_mlp_wmma_kernel(const float* __restrict__ feat,
                          const float* __restrict__ geom,
                          const float* __restrict__ W1,
                          const float* __restrict__ b1,
                          const float* __restrict__ W2,
                          const float* __restrict__ b2,
                          float* __restrict__ out)
{
    __shared__ float A[16 * ASTRIDE];   // combined inputs, cols 0..71 (70,71 zero-pad)
    __shared__ float Hs[16 * HSTRIDE];  // hidden activations

    const int tid     = threadIdx.x;
    const int rowBase = blockIdx.x * 16;

    // stage features (16 x 64)
    for (int idx = tid; idx < 16 * FEAT_DIM; idx += 256) {
        int r = idx >> 6, c = idx & 63;
        A[r * ASTRIDE + c] = feat[(size_t)(rowBase + r) * FEAT_DIM + c];
    }
    // stage geometry (16 x 6) + zero-pad K to 72
    for (int idx = tid; idx < 16 * 8; idx += 256) {
        int r = idx >> 3, c = idx & 7;
        A[r * ASTRIDE + FEAT_DIM + c] =
            (c < 6) ? geom[(size_t)(rowBase + r) * 6 + c] : 0.0f;
    }
    __syncthreads();

    const int wave = tid >> 5;          // 0..7 -> output column slab
    const int lane = tid & 31;
    const int hi   = lane >> 4;         // lanes 16-31 carry K+2/K+3 (A) / M+8 (C)
    const int m    = lane & 15;
    const int col  = wave * 16 + m;

    // ---- layer 1: C[16x16] += A[16x4] * W1[4x16], K = 0..71 step 4 ----
    v8f c1 = {};
    for (int k = 0; k < KPAD; k += 4) {
        int ka = k + 2 * hi;
        v2f a; a.x = A[m * ASTRIDE + ka];
               a.y = A[m * ASTRIDE + ka + 1];
        v2f bv;
        bv.x = (ka     < DIN) ? W1[(size_t)ka       * HID + col] : 0.0f;
        bv.y = (ka + 1 < DIN) ? W1[(size_t)(ka + 1) * HID + col] : 0.0f;
        c1 = __builtin_amdgcn_wmma_f32_16x16x4_f32(
                 false, a, false, bv, (short)0, c1, false, false);
    }
    {
        float bias = b1[col];
#pragma unroll
        for (int r = 0; r < 8; ++r) {
            int row = r + 8 * hi;
            Hs[row * HSTRIDE + col] = fmaxf(c1[r] + bias, 0.0f);
        }
    }
    __syncthreads();

    // ---- layer 2: K = 128, no padding needed ----
    v8f c2 = {};
    for (int k = 0; k < HID; k += 4) {
        int ka = k + 2 * hi;
        v2f a; a.x = Hs[m * HSTRIDE + ka];
               a.y = Hs[m * HSTRIDE + ka + 1];
        v2f bv; bv.x = W2[(size_t)ka       * HID + col];
                bv.y = W2[(size_t)(ka + 1) * HID + col];
        c2 = __builtin_amdgcn_wmma_f32_16x16x4_f32(
                 false, a, false, bv, (short)0, c2, false, false);
    }
    {
        float bias = b2[col];
#pragma unroll
        for (int r = 0; r < 8; ++r) {
            int row = r + 8 * hi;
            out[(size_t)(rowBase + row) * HID + col] = fmaxf(c2[r] + bias, 0.0f);
        }
    }
}

// =====================================================================
extern "C" void kernel_launch(void* const* d_in, const int* in_sizes, int n_in,
                              void* d_out, int out_size, void* d_ws, size_t ws_size,
                              hipStream_t stream)
{
    const float*         points = (const float*)d_in[0];          // (B,N,3)
    const float*         feat   = (const float*)d_in[1];          // (B,N,64)
    const unsigned char* mask   = (const unsigned char*)d_in[2];  // (B,N) bool
    const float*         W1     = (const float*)d_in[3];          // (70,128)
    const float*         b1     = (const float*)d_in[4];          // (128,)
    const float*         W2     = (const float*)d_in[5];          // (128,128)
    const float*         b2     = (const float*)d_in[6];          // (128,)
    float*               out    = (float*)d_out;                  // (B,N,128)

    // workspace layout: [0,256) centroid sums, [256, ...) geom (B*N*6 f32)
    float* center_ws = (float*)d_ws;
    float* geom_ws   = (float*)((char*)d_ws + 256);

    leaf_center_kernel<<<dim3(BATCH), dim3(256), 0, stream>>>(points, mask, center_ws);

    leaf_geom_kernel<<<dim3(NPTS / GPTS, BATCH), dim3(256), 0, stream>>>(
        points, mask, center_ws, geom_ws);

    leaf_mlp_wmma_kernel<<<dim3((BATCH * NPTS) / 16), dim3(256), 0, stream>>>(
        feat, geom_ws, W1, b1, W2, b2, out);
}